// EntmaxScaledDotProduct_37409165148926
// MI455X (gfx1250) — compile-verified
//
#include <hip/hip_runtime.h>
#include <hip/hip_bf16.h>

typedef __attribute__((ext_vector_type(2))) float v2f;
typedef __attribute__((ext_vector_type(8))) float v8f;

namespace {
constexpr int kB = 16;
constexpr int kN = 2048;
constexpr int kD = 64;
constexpr int kTileRows = 16;            // one WMMA M-tile per workgroup
constexpr int kStride = kN + 2;          // LDS row stride (bank-conflict pad)
constexpr int kWaves = 8;
constexpr float kInvAm1 = 1.0f / 0.3f;   // 1/(alpha-1), alpha=1.3
constexpr int kNIter = 50;
}

__device__ __forceinline__ float entmax_pow(float t) {
  // t^(10/3) for t>0 via v_log_f32 / v_exp_f32; caller selects away t<=0.
  return __builtin_amdgcn_exp2f(kInvAm1 * __builtin_amdgcn_logf(t));
}

__global__ __launch_bounds__(256) void entmax_attn_kernel(
    const float* __restrict__ q, const float* __restrict__ k,
    const float* __restrict__ v, float* __restrict__ outp,
    float* __restrict__ attnp) {
  __shared__ float smem[kTileRows * kStride];   // score row block -> p -> attn
  __shared__ float spart[4 * 16 * 16];          // phase-4 K-split partials

  const int tid = threadIdx.x;
  const int lane = tid & 31;
  const int wv = tid >> 5;
  const int bIdx = blockIdx.x >> 7;             // 128 row-tiles per batch
  const int rowBase = (blockIdx.x & 127) * kTileRows;

  const float* qb = q + (size_t)bIdx * kN * kD;
  const float* kb = k + (size_t)bIdx * kN * kD;
  const float* vb = v + (size_t)bIdx * kN * kD;

  const int mrow = lane & 15;                   // M (or N) index within fragment
  const int kOff = (lane >> 4) * 2;             // K sub-offset within 4-chunk

  // ---------------- Phase 1: scores = (Q K^T)/8 via V_WMMA_F32_16X16X4_F32 ----
  v2f afrag[16];
#pragma unroll
  for (int kc = 0; kc < 16; ++kc) {
    v2f a = *(const v2f*)(qb + (rowBase + mrow) * kD + kc * 4 + kOff);
    afrag[kc].x = a.x * 0.125f;                 // exact pow2 scale: 1/sqrt(64)
    afrag[kc].y = a.y * 0.125f;
  }

  for (int ct = wv; ct < kN / 16; ct += kWaves) {
    v8f c = {};
    const float* bp = kb + (ct * 16 + mrow) * kD + kOff;
#pragma unroll
    for (int kc = 0; kc < 16; ++kc) {
      v2f bf = *(const v2f*)(bp + kc * 4);
      c = __builtin_amdgcn_wmma_f32_16x16x4_f32(false, afrag[kc], false, bf,
                                                (short)0, c, false, false);
    }
#pragma unroll
    for (int r = 0; r < 8; ++r) {
      int m = r + (lane >> 4) * 8;
      smem[m * kStride + ct * 16 + mrow] = c[r];
    }
  }
  __syncthreads();

  // ---------------- Phase 2: per-row stats + 50-step bisection (2 rows/wave) --
  const int r0 = 2 * wv, r1 = 2 * wv + 1;
  float* p0 = &smem[r0 * kStride];              // 8B-aligned (stride even)
  float* p1 = &smem[r1 * kStride];
  const v2f* q0 = (const v2f*)p0;
  const v2f* q1 = (const v2f*)p1;

  float mx0 = -3.4e38f, mx1 = -3.4e38f, mn0 = 3.4e38f, mn1 = 3.4e38f;
  for (int i = 0; i < kN / 64; ++i) {
    v2f e0 = q0[lane + 32 * i];
    v2f e1 = q1[lane + 32 * i];
    mx0 = fmaxf(mx0, fmaxf(e0.x, e0.y)); mn0 = fminf(mn0, fminf(e0.x, e0.y));
    mx1 = fmaxf(mx1, fmaxf(e1.x, e1.y)); mn1 = fminf(mn1, fminf(e1.x, e1.y));
  }
#pragma unroll
  for (int off = 16; off; off >>= 1) {
    mx0 = fmaxf(mx0, __shfl_xor(mx0, off, 32));
    mn0 = fminf(mn0, __shfl_xor(mn0, off, 32));
    mx1 = fmaxf(mx1, __shfl_xor(mx1, off, 32));
    mn1 = fminf(mn1, __shfl_xor(mn1, off, 32));
  }

  // x = s - mx; tau_lo = min(x)-1, tau_hi = max(x)-1e-6 = -1e-6
  float lo0 = (mn0 - mx0) - 1.0f, hi0 = -1e-6f;
  float lo1 = (mn1 - mx1) - 1.0f, hi1 = -1e-6f;

  for (int it = 0; it < kNIter; ++it) {
    float tau0 = 0.5f * (lo0 + hi0), tau1 = 0.5f * (lo1 + hi1);
    float tt0 = mx0 + tau0, tt1 = mx1 + tau1;
    float s0 = 0.f, s1 = 0.f;
    for (int i = 0; i < kN / 64; ++i) {
      v2f e0 = q0[lane + 32 * i];
      v2f e1 = q1[lane + 32 * i];
      float t0a = e0.x - tt0, t0b = e0.y - tt0;
      float t1a = e1.x - tt1, t1b = e1.y - tt1;
      s0 += (t0a > 0.f) ? entmax_pow(t0a) : 0.f;
      s0 += (t0b > 0.f) ? entmax_pow(t0b) : 0.f;
      s1 += (t1a > 0.f) ? entmax_pow(t1a) : 0.f;
      s1 += (t1b > 0.f) ? entmax_pow(t1b) : 0.f;
    }
#pragma unroll
    for (int off = 16; off; off >>= 1) {
      s0 += __shfl_xor(s0, off, 32);
      s1 += __shfl_xor(s1, off, 32);
    }
    // f = sum^0.3 > 1  <=>  sum > 1  (monotone, sum >= 0)
    bool big0 = s0 > 1.0f, big1 = s1 > 1.0f;
    lo0 = big0 ? tau0 : lo0; hi0 = big0 ? hi0 : tau0;
    lo1 = big1 ? tau1 : lo1; hi1 = big1 ? hi1 : tau1;
  }

  // ---------------- Phase 3: final p, normalize, stream attn to HBM (NT) -----
  {
    float tt0 = mx0 + 0.5f * (lo0 + hi0);
    float tt1 = mx1 + 0.5f * (lo1 + hi1);
    float s0 = 0.f, s1 = 0.f;
    v2f* w0 = (v2f*)p0;
    v2f* w1 = (v2f*)p1;
    for (int i = 0; i < kN / 64; ++i) {
      v2f e0 = w0[lane + 32 * i];
      v2f e1 = w1[lane + 32 * i];
      float t0a = e0.x - tt0, t0b = e0.y - tt0;
      float t1a = e1.x - tt1, t1b = e1.y - tt1;
      v2f o0, o1;
      o0.x = (t0a > 0.f) ? entmax_pow(t0a) : 0.f;
      o0.y = (t0b > 0.f) ? entmax_pow(t0b) : 0.f;
      o1.x = (t1a > 0.f) ? entmax_pow(t1a) : 0.f;
      o1.y = (t1b > 0.f) ? entmax_pow(t1b) : 0.f;
      w0[lane + 32 * i] = o0;
      w1[lane + 32 * i] = o1;
      s0 += o0.x + o0.y;
      s1 += o1.x + o1.y;
    }
#pragma unroll
    for (int off = 16; off; off >>= 1) {
      s0 += __shfl_xor(s0, off, 32);
      s1 += __shfl_xor(s1, off, 32);
    }
    float rn0 = 1.0f / (s0 + 1e-12f);
    float rn1 = 1.0f / (s1 + 1e-12f);
    v2f* g0 = (v2f*)(attnp + ((size_t)bIdx * kN + rowBase + r0) * kN);
    v2f* g1 = (v2f*)(attnp + ((size_t)bIdx * kN + rowBase + r1) * kN);
    for (int i = 0; i < kN / 64; ++i) {
      v2f a0 = w0[lane + 32 * i];
      v2f a1 = w1[lane + 32 * i];
      a0.x *= rn0; a0.y *= rn0;
      a1.x *= rn1; a1.y *= rn1;
      w0[lane + 32 * i] = a0;                   // keep normalized attn for P4
      w1[lane + 32 * i] = a1;
      // 268MB stream, never re-read: bypass L2 (TH_NT) to protect K/V reuse
      __builtin_nontemporal_store(a0, g0 + lane + 32 * i);
      __builtin_nontemporal_store(a1, g1 + lane + 32 * i);
    }
  }
  __syncthreads();

  // ---------------- Phase 4: out = attn @ V via WMMA (4 N-tiles x 2 K-halves)-
  {
    const int ntile = wv & 3;
    const int khalf = wv >> 2;
    v8f acc = {};
    const float* vcol = vb + ntile * 16 + mrow;  // column of V for this lane
    const int k0base = khalf * (kN / 2);
    for (int kk = 0; kk < kN / 2; kk += 4) {
      int kidx = k0base + kk + kOff;
      v2f a = *(const v2f*)&smem[mrow * kStride + kidx];
      v2f bf;
      bf.x = vcol[(size_t)kidx * kD];
      bf.y = vcol[(size_t)(kidx + 1) * kD];
      acc = __builtin_amdgcn_wmma_f32_16x16x4_f32(false, a, false, bf,
                                                  (short)0, acc, false, false);
    }
    if (khalf == 1) {
#pragma unroll
      for (int r = 0; r < 8; ++r) {
        int rr = r + (lane >> 4) * 8;
        spart[ntile * 256 + rr * 16 + mrow] = acc[r];
      }
    }
    __syncthreads();
    if (khalf == 0) {
      float* ob = outp + ((size_t)bIdx * kN + rowBase) * kD;
#pragma unroll
      for (int r = 0; r < 8; ++r) {
        int rr = r + (lane >> 4) * 8;
        ob[rr * kD + ntile * 16 + mrow] =
            acc[r] + spart[ntile * 256 + rr * 16 + mrow];
      }
    }
  }
}

extern "C" void kernel_launch(void* const* d_in, const int* in_sizes, int n_in,
                              void* d_out, int out_size, void* d_ws,
                              size_t ws_size, hipStream_t stream) {
  (void)in_sizes; (void)n_in; (void)out_size; (void)d_ws; (void)ws_size;
  const float* q = (const float*)d_in[0];
  const float* k = (const float*)d_in[1];
  const float* v = (const float*)d_in[2];
  float* outp = (float*)d_out;                       // [B,N,D] first
  float* attnp = outp + (size_t)kB * kN * kD;        // then [B,N,N]
  dim3 grid(kB * (kN / kTileRows));                  // 2048 workgroups
  entmax_attn_kernel<<<grid, 256, 0, stream>>>(q, k, v, outp, attnp);
}